// Trellis4DGS4DCanonical_75093208203466
// MI455X (gfx1250) — compile-verified
//
#include <hip/hip_runtime.h>

// ---------------------------------------------------------------------------
// Trellis4DGS warp pipeline for MI455X (gfx1250, wave32).
//  - top-k soft-assign : VALU + LDS (selection-bound, WMMA not applicable)
//  - MLP GEMMs         : V_WMMA_F32_16X16X4_F32 (full f32 precision)
//  - voxel thinning    : hash table w/ ordered-uint atomics (deterministic)
// ---------------------------------------------------------------------------

#define NPTS 262144
#define NANC 4096
#define TOPK 8
#define DDIM 128
#define HDIM 256
#define NT   8
#define TS   (1u << 20)   // hash table slots

typedef float v2f __attribute__((ext_vector_type(2)));
typedef float v8f __attribute__((ext_vector_type(8)));

// monotone float <-> uint encoding (for atomic min/max on floats)
__device__ __forceinline__ unsigned fenc(float f) {
  unsigned b = __float_as_uint(f);
  return (b & 0x80000000u) ? ~b : (b | 0x80000000u);
}
__device__ __forceinline__ float fdec(unsigned k) {
  unsigned b = (k & 0x80000000u) ? (k ^ 0x80000000u) : ~k;
  return __uint_as_float(b);
}
__device__ __forceinline__ float silu(float x) {
  return x / (1.0f + __expf(-x));
}

// ---------------------------------------------------------------- init tables
__global__ void k_init(unsigned* __restrict__ keys, unsigned* __restrict__ maxb,
                       unsigned* __restrict__ winner, unsigned* __restrict__ minenc) {
  int i = blockIdx.x * blockDim.x + threadIdx.x;
  int stride = gridDim.x * blockDim.x;
  for (unsigned s = i; s < TS; s += stride) {
    keys[s]   = 0xFFFFFFFFu;   // empty sentinel
    maxb[s]   = 0u;            // encoded -inf-ish (lowest key)
    winner[s] = 0xFFFFFFFFu;
  }
  if (i < 3) minenc[i] = 0xFFFFFFFFu;  // encoded +inf region
}

// ------------------------------------------------------------ global xyz min
__global__ __launch_bounds__(256)
void k_min(const float* __restrict__ xyz, unsigned* __restrict__ minenc, int n) {
  __shared__ float sm[3 * 256];
  int tid = threadIdx.x;
  float mx = 3.0e38f, my = 3.0e38f, mz = 3.0e38f;
  for (int i = blockIdx.x * blockDim.x + tid; i < n; i += gridDim.x * blockDim.x) {
    mx = fminf(mx, xyz[3 * i + 0]);
    my = fminf(my, xyz[3 * i + 1]);
    mz = fminf(mz, xyz[3 * i + 2]);
  }
  sm[tid] = mx; sm[256 + tid] = my; sm[512 + tid] = mz;
  __syncthreads();
  for (int s = 128; s > 0; s >>= 1) {
    if (tid < s) {
      sm[tid]       = fminf(sm[tid],       sm[tid + s]);
      sm[256 + tid] = fminf(sm[256 + tid], sm[256 + tid + s]);
      sm[512 + tid] = fminf(sm[512 + tid], sm[512 + tid + s]);
    }
    __syncthreads();
  }
  if (tid == 0) {
    atomicMin(&minenc[0], fenc(sm[0]));
    atomicMin(&minenc[1], fenc(sm[256]));
    atomicMin(&minenc[2], fenc(sm[512]));
  }
}

// ------------------------------------------- pass 1: insert + per-voxel max
__global__ __launch_bounds__(256)
void k_hash1(const float* __restrict__ xyz, const float* __restrict__ score,
             const float* __restrict__ noise, const unsigned* __restrict__ minenc,
             unsigned* __restrict__ keys, unsigned* __restrict__ maxb,
             unsigned* __restrict__ slotOf) {
  int i = blockIdx.x * blockDim.x + threadIdx.x;
  if (i >= NPTS) return;
  float mnx = fdec(minenc[0]), mny = fdec(minenc[1]), mnz = fdec(minenc[2]);
  int vx = (int)floorf((xyz[3 * i + 0] - mnx) / 0.05f);
  int vy = (int)floorf((xyz[3 * i + 1] - mny) / 0.05f);
  int vz = (int)floorf((xyz[3 * i + 2] - mnz) / 0.05f);
  unsigned h = ((unsigned)vx * 73856093u) ^ ((unsigned)vy * 19349663u) ^
               ((unsigned)vz * 83492791u);
  unsigned slot = (h * 2654435761u) & (TS - 1u);
  for (;;) {   // linear probing (h == 0xFFFFFFFF sentinel collision: negligible)
    unsigned prev = atomicCAS(&keys[slot], 0xFFFFFFFFu, h);
    if (prev == 0xFFFFFFFFu || prev == h) break;
    slot = (slot + 1u) & (TS - 1u);
  }
  slotOf[i] = slot;
  float s2 = score[i] + 1e-6f * noise[i];
  atomicMax(&maxb[slot], fenc(s2));
}

// ----------------------- pass 2: min index among eligible (score2>=max-1e-12)
__global__ __launch_bounds__(256)
void k_hash2(const float* __restrict__ score, const float* __restrict__ noise,
             const unsigned* __restrict__ slotOf, const unsigned* __restrict__ maxb,
             unsigned* __restrict__ winner) {
  int i = blockIdx.x * blockDim.x + threadIdx.x;
  if (i >= NPTS) return;
  unsigned slot = slotOf[i];
  float s2 = score[i] + 1e-6f * noise[i];
  float mx = fdec(maxb[slot]);
  if (s2 >= mx - 1e-12f) atomicMin(&winner[slot], (unsigned)i);
}

// ---------------------------------------------------------- pass 3: keep mask
__global__ __launch_bounds__(256)
void k_keep(const unsigned* __restrict__ slotOf, const unsigned* __restrict__ winner,
            float* __restrict__ keep) {
  int i = blockIdx.x * blockDim.x + threadIdx.x;
  if (i >= NPTS) return;
  keep[i] = (winner[slotOf[i]] == (unsigned)i) ? 1.0f : 0.0f;
}

// ----------------------------- tiny MLP head: gate = silu(silu(temb@W1)@W2)
__global__ __launch_bounds__(256)
void k_gate(const int* __restrict__ t_ids, const float* __restrict__ W1,
            const float* __restrict__ b1, const float* __restrict__ W2,
            const float* __restrict__ b2, float* __restrict__ gate) {
  __shared__ float temb[NT * 32];
  __shared__ float h1[NT * HDIM];
  int tid = threadIdx.x;
  float tmin = 3.0e38f, tmax = -3.0e38f;
  for (int i = 0; i < NT; i++) {
    float tv = (float)t_ids[i];
    tmin = fminf(tmin, tv);
    tmax = fmaxf(tmax, tv);
  }
  tmax = fmaxf(tmax, tmin + 1.0f);
  {
    int t = tid >> 5;            // 0..7
    int e = tid & 31;            // 0..31
    float tn = ((float)t_ids[t] - tmin) / (tmax - tmin);
    int j = e & 15;
    float fr = __expf((8.0f / 15.0f) * (float)j);   // linspace(0,8,16) -> exp
    float ph = tn * fr;
    temb[t * 32 + e] = (e < 16) ? __sinf(ph) : __cosf(ph);
  }
  __syncthreads();
  for (int o = tid; o < NT * HDIM; o += 256) {
    int t = o >> 8, j = o & 255;
    float acc = b1[j];
    for (int e = 0; e < 32; e++) acc += temb[t * 32 + e] * W1[e * HDIM + j];
    h1[o] = silu(acc);
  }
  __syncthreads();
  for (int o = tid; o < NT * HDIM; o += 256) {
    int t = o >> 8, j = o & 255;
    float acc = b2[j];
    for (int hh = 0; hh < HDIM; hh++) acc += h1[t * HDIM + hh] * W2[hh * HDIM + j];
    gate[o] = silu(acc);
  }
}

// --------------------------------------------- z_h = z @ Wz + bz  (f32 WMMA)
// one wave per 16x16 tile; K-loop over D=128 in steps of 4
__global__ __launch_bounds__(32)
void k_zh(const float* __restrict__ z, const float* __restrict__ Wz,
          const float* __restrict__ bz, float* __restrict__ zh) {
  int tile = blockIdx.x;                // 256 row-tiles * 16 col-tiles
  int a0 = (tile >> 4) * 16;
  int n0 = (tile & 15) * 16;
  int lane = threadIdx.x;
  int l15  = lane & 15;
  int koff = (lane >> 4) * 2;           // lanes 16..31 carry K=2,3
  const float* zr = z + (size_t)(a0 + l15) * DDIM;
  int n = n0 + l15;
  v8f c = {};
  for (int k = 0; k < DDIM; k += 4) {
    v2f av, bv;
    av.x = zr[k + koff];
    av.y = zr[k + koff + 1];
    bv.x = Wz[(size_t)(k + koff) * HDIM + n];
    bv.y = Wz[(size_t)(k + koff + 1) * HDIM + n];
    c = __builtin_amdgcn_wmma_f32_16x16x4_f32(false, av, false, bv,
                                              (short)0, c, false, false);
  }
  float bn = bz[n];
  int mbase = (lane >> 4) * 8;          // D rows: lanes<16 -> M=0..7, else 8..15
#pragma unroll
  for (int m = 0; m < 8; m++)
    zh[(size_t)(a0 + mbase + m) * HDIM + n] = c[m] + bn;
}

// ------------- deltas = silu((gate*zh) @ W3 + b3) @ W4 + b4  (f32 WMMA, fused)
// one wave per 16 fused rows; loops over all 16 N-tiles, contracts W4 in regs
__global__ __launch_bounds__(32)
void k_mlp2(const float* __restrict__ zh, const float* __restrict__ gate,
            const float* __restrict__ W3, const float* __restrict__ b3,
            const float* __restrict__ W4, const float* __restrict__ b4,
            float* __restrict__ deltas) {
  const int lane = threadIdx.x;
  const int r0 = blockIdx.x * 16;       // fused row tile (t*4096 + a)
  const int t  = r0 >> 12;              // 4096 rows per t, tiles never straddle
  const int a0 = r0 & (NANC - 1);
  const int half = lane >> 4;
  const int l15  = lane & 15;
  const int koff = half * 2;
  const float* zrow = zh + (size_t)(a0 + l15) * HDIM;
  const float* grow = gate + (size_t)t * HDIM;

  float dacc[8][3];
#pragma unroll
  for (int m = 0; m < 8; m++)
#pragma unroll
    for (int cc = 0; cc < 3; cc++) dacc[m][cc] = 0.0f;

#pragma unroll 1
  for (int j = 0; j < 16; ++j) {        // N-tiles of h3
    const int n = j * 16 + l15;
    v8f c = {};
    for (int k = 0; k < HDIM; k += 4) {
      v2f av, bv;
      av.x = zrow[k + koff]     * grow[k + koff];
      av.y = zrow[k + koff + 1] * grow[k + koff + 1];
      bv.x = W3[(size_t)(k + koff) * HDIM + n];
      bv.y = W3[(size_t)(k + koff + 1) * HDIM + n];
      c = __builtin_amdgcn_wmma_f32_16x16x4_f32(false, av, false, bv,
                                                (short)0, c, false, false);
    }
    float b3n = b3[n];
    float w40 = W4[n * 3 + 0], w41 = W4[n * 3 + 1], w42 = W4[n * 3 + 2];
#pragma unroll
    for (int m = 0; m < 8; m++) {
      float s = silu(c[m] + b3n);
      dacc[m][0] += s * w40;
      dacc[m][1] += s * w41;
      dacc[m][2] += s * w42;
    }
  }
  // reduce over the 16 lanes of each half (each lane held one N column)
#pragma unroll
  for (int m = 0; m < 8; m++)
#pragma unroll
    for (int cc = 0; cc < 3; cc++) {
      float v = dacc[m][cc];
      v += __shfl_xor(v, 1, 16);
      v += __shfl_xor(v, 2, 16);
      v += __shfl_xor(v, 4, 16);
      v += __shfl_xor(v, 8, 16);
      dacc[m][cc] = v;
    }
  if (l15 == 0) {                       // lanes 0 (M=0..7) and 16 (M=8..15)
    int mbase = half * 8;
#pragma unroll
    for (int m = 0; m < 8; m++) {
      float* dst = deltas + ((size_t)t * NANC + a0 + mbase + m) * 3;
      dst[0] = dacc[m][0] + b4[0];
      dst[1] = dacc[m][1] + b4[1];
      dst[2] = dacc[m][2] + b4[2];
    }
  }
}

// -------- top-8 soft assign + warp + mask.  LDS: anchors+|a|^2, then deltas_t
__global__ __launch_bounds__(256)
void k_warp(const float* __restrict__ xyz, const float* __restrict__ anchors,
            const float* __restrict__ deltas, const float* __restrict__ keep,
            float* __restrict__ out) {
  __shared__ float smem[16384];         // 64 KB
  const int tid = threadIdx.x;
  const int n = blockIdx.x * 256 + tid;
  for (int i = tid; i < NANC * 3; i += 256) smem[i] = anchors[i];
  __syncthreads();
  for (int i = tid; i < NANC; i += 256) {
    float ax = smem[3 * i], ay = smem[3 * i + 1], az = smem[3 * i + 2];
    smem[NANC * 3 + i] = ax * ax + ay * ay + az * az;
  }
  __syncthreads();

  float px = xyz[3 * n], py = xyz[3 * n + 1], pz = xyz[3 * n + 2];
  float pp = px * px + py * py + pz * pz;
  float d[TOPK]; int id[TOPK];
#pragma unroll
  for (int j = 0; j < TOPK; j++) { d[j] = 3.4e38f; id[j] = 0; }
  float dmax = 3.4e38f; int imax = 0;
  for (int a = 0; a < NANC; a++) {      // uniform a -> LDS broadcast reads
    float ax = smem[3 * a], ay = smem[3 * a + 1], az = smem[3 * a + 2];
    float dot = px * ax + py * ay + pz * az;
    float d2 = fmaxf(pp + smem[NANC * 3 + a] - 2.0f * dot, 0.0f);
    if (d2 < dmax) {                    // strict <: ties keep earlier index
      d[imax] = d2; id[imax] = a;
      dmax = d[0]; imax = 0;
#pragma unroll
      for (int j = 1; j < TOPK; j++)
        if (d[j] > dmax) { dmax = d[j]; imax = j; }
    }
  }
  // softmax over -d2/(2*sigma^2); permutation-invariant for the weighted sum
  float dmin = d[0];
#pragma unroll
  for (int j = 1; j < TOPK; j++) dmin = fminf(dmin, d[j]);
  float wk[TOPK]; float sum = 0.0f;
#pragma unroll
  for (int j = 0; j < TOPK; j++) { wk[j] = __expf((dmin - d[j]) * 200.0f); sum += wk[j]; }
  float inv = 1.0f / sum;
#pragma unroll
  for (int j = 0; j < TOPK; j++) wk[j] *= inv;

  float kp = keep[n];
  for (int t = 0; t < NT; t++) {
    __syncthreads();                    // previous LDS use done
    for (int i = tid; i < NANC * 3; i += 256) smem[i] = deltas[t * NANC * 3 + i];
    __syncthreads();
    float ox = px, oy = py, oz = pz;
#pragma unroll
    for (int j = 0; j < TOPK; j++) {
      int a3 = id[j] * 3; float w = wk[j];
      ox += w * smem[a3]; oy += w * smem[a3 + 1]; oz += w * smem[a3 + 2];
    }
    size_t o = ((size_t)t * NPTS + n) * 3;
    out[o] = ox * kp; out[o + 1] = oy * kp; out[o + 2] = oz * kp;
  }
}

// ---------------------------------------------------------------------------
extern "C" void kernel_launch(void* const* d_in, const int* in_sizes, int n_in,
                              void* d_out, int out_size, void* d_ws, size_t ws_size,
                              hipStream_t stream) {
  const float* xyz     = (const float*)d_in[0];
  const float* anchors = (const float*)d_in[1];
  const float* z       = (const float*)d_in[2];
  const float* score   = (const float*)d_in[3];
  const float* noise   = (const float*)d_in[4];
  const int*   t_ids   = (const int*)d_in[5];
  const float* Wz = (const float*)d_in[6];
  const float* bz = (const float*)d_in[7];
  const float* W1 = (const float*)d_in[8];
  const float* b1 = (const float*)d_in[9];
  const float* W2 = (const float*)d_in[10];
  const float* b2 = (const float*)d_in[11];
  const float* W3 = (const float*)d_in[12];
  const float* b3 = (const float*)d_in[13];
  const float* W4 = (const float*)d_in[14];
  const float* b4 = (const float*)d_in[15];
  float* out = (float*)d_out;

  // workspace carve-out (~18.5 MB), 256B aligned
  char* p = (char*)d_ws;
  auto carve = [&](size_t bytes) -> char* {
    char* r = p;
    p += (bytes + 255) & ~(size_t)255;
    return r;
  };
  float*    gate   = (float*)carve((size_t)NT * HDIM * 4);
  float*    zh     = (float*)carve((size_t)NANC * HDIM * 4);
  float*    deltas = (float*)carve((size_t)NT * NANC * 3 * 4);
  unsigned* minenc = (unsigned*)carve(3 * 4);
  unsigned* keys   = (unsigned*)carve((size_t)TS * 4);
  unsigned* maxb   = (unsigned*)carve((size_t)TS * 4);
  unsigned* winner = (unsigned*)carve((size_t)TS * 4);
  unsigned* slotOf = (unsigned*)carve((size_t)NPTS * 4);
  float*    keep   = (float*)carve((size_t)NPTS * 4);

  const int NB = NPTS / 256;   // 1024

  k_init <<<1024, 256, 0, stream>>>(keys, maxb, winner, minenc);
  k_min  <<<256, 256, 0, stream>>>(xyz, minenc, NPTS);
  k_hash1<<<NB, 256, 0, stream>>>(xyz, score, noise, minenc, keys, maxb, slotOf);
  k_hash2<<<NB, 256, 0, stream>>>(score, noise, slotOf, maxb, winner);
  k_keep <<<NB, 256, 0, stream>>>(slotOf, winner, keep);
  k_gate <<<1, 256, 0, stream>>>(t_ids, W1, b1, W2, b2, gate);
  k_zh   <<<(NANC / 16) * (HDIM / 16), 32, 0, stream>>>(z, Wz, bz, zh);
  k_mlp2 <<<(NT * NANC) / 16, 32, 0, stream>>>(zh, gate, W3, b3, W4, b4, deltas);
  k_warp <<<NB, 256, 0, stream>>>(xyz, anchors, deltas, keep, out);
}